// MultiHeadLTC_34763465294530
// MI455X (gfx1250) — compile-verified
//
#include <hip/hip_runtime.h>
#include <hip/hip_bf16.h>

#define V_ 8
#define B_ 512
#define T_ 64
#define U_ 64
#define H_ 256
#define C_ 10
#define UNFOLDS_ 6
#define EPS_ 1e-8f
#define LOG2E_ 1.4426950408889634f

typedef float v2f_t __attribute__((ext_vector_type(2)));
typedef float v8f_t __attribute__((ext_vector_type(8)));

__device__ __forceinline__ float softplusf(float x) {
    return (x > 20.0f) ? x : log1pf(expf(x));
}

// ---------------------------------------------------------------------------
// Prep 1: fold sigma/mu/w/erev into interleaved float4 {sigma*log2e,
// sigma*mu*log2e, softplus(w)*erev, softplus(w)} per (v,j,k).
// ---------------------------------------------------------------------------
__global__ void ltc_prep_mat(const float* __restrict__ w,
                             const float* __restrict__ sigma,
                             const float* __restrict__ mu,
                             const float* __restrict__ erev,
                             float* __restrict__ PK) {
    int idx = blockIdx.x * blockDim.x + threadIdx.x;
    if (idx >= V_ * U_ * U_) return;
    float sg = sigma[idx];
    float wp = softplusf(w[idx]);
    float4 p;
    p.x = sg * LOG2E_;
    p.y = sg * mu[idx] * LOG2E_;
    p.z = wp * erev[idx];
    p.w = wp;
    ((float4*)PK)[idx] = p;
}

// ---------------------------------------------------------------------------
// Prep 2: per-(v,u) vectors. Sensory arrays are [V,1,U] -> flat [V*U] (I==1).
// ---------------------------------------------------------------------------
__global__ void ltc_prep_vec(const float* __restrict__ gleak,
                             const float* __restrict__ vleak,
                             const float* __restrict__ cm,
                             const float* __restrict__ sw,
                             const float* __restrict__ ssig,
                             const float* __restrict__ smu,
                             const float* __restrict__ serev,
                             float* __restrict__ vecs) {
    int idx = blockIdx.x * blockDim.x + threadIdx.x;
    const int VU = V_ * U_;
    if (idx >= VU) return;
    float gl  = softplusf(gleak[idx]);
    float cmt = softplusf(cm[idx]) * (float)UNFOLDS_;
    float swp = softplusf(sw[idx]);
    vecs[0 * VU + idx] = cmt;                         // cm_t
    vecs[1 * VU + idx] = gl * vleak[idx];             // gl*vleak
    vecs[2 * VU + idx] = gl;                          // gl
    vecs[3 * VU + idx] = ssig[idx] * LOG2E_;          // sensory sigma * log2e
    vecs[4 * VU + idx] = ssig[idx] * smu[idx] * LOG2E_;
    vecs[5 * VU + idx] = swp * serev[idx];            // sw_p * serev
    vecs[6 * VU + idx] = swp;                         // sw_p
}

// ---------------------------------------------------------------------------
// Main LTC scan. One wave per (head, batch element). Lane owns k = lane and
// k = lane+32. v-state per wave lives in LDS (per-wave region, no barriers
// needed after staging: LDS ops from the same wave stay in order).
// Dynamic LDS: [U*U float4 params][8 waves * U floats of v].
// ---------------------------------------------------------------------------
__global__ void __launch_bounds__(256, 1)
ltc_scan(const float* __restrict__ x,        // [V,B,T] (I==1)
         const float* __restrict__ PK,       // [V,U,U,4]
         const float* __restrict__ vecs,     // [7,V*U]
         const float* __restrict__ input_w,  // [V]
         const float* __restrict__ input_b,  // [V]
         const float* __restrict__ output_w, // [V,U]
         const float* __restrict__ output_b, // [V,U]
         float* __restrict__ feats)          // [B, V*U]
{
    extern __shared__ float smem[];
    float4* sPK4 = (float4*)smem;           // U*U float4 = 64KB
    float*  sV   = smem + U_ * U_ * 4;      // 8 * U floats

    const int v     = blockIdx.x >> 6;      // B_/8 == 64 batch tiles per head
    const int btile = blockIdx.x & 63;
    const int wid   = threadIdx.x >> 5;
    const int lane  = threadIdx.x & 31;
    const int b     = btile * 8 + wid;
    const int k0    = lane, k1 = lane + 32;

    // Stage the head's packed parameter matrix into LDS.
    const float4* gPK = (const float4*)(PK + (size_t)v * U_ * U_ * 4);
    for (int i = threadIdx.x; i < U_ * U_; i += 256) sPK4[i] = gPK[i];

    // Per-lane constants (tiny, straight from L2).
    const int VU = V_ * U_;
    const int p0 = v * U_ + k0, p1 = v * U_ + k1;
    const float cmt0 = vecs[0*VU+p0], cmt1 = vecs[0*VU+p1];
    const float glv0 = vecs[1*VU+p0], glv1 = vecs[1*VU+p1];
    const float gl0  = vecs[2*VU+p0], gl1  = vecs[2*VU+p1];
    const float ssa0 = vecs[3*VU+p0], ssa1 = vecs[3*VU+p1];
    const float ssc0 = vecs[4*VU+p0], ssc1 = vecs[4*VU+p1];
    const float swe0 = vecs[5*VU+p0], swe1 = vecs[5*VU+p1];
    const float swp0 = vecs[6*VU+p0], swp1 = vecs[6*VU+p1];
    const float iw = input_w[v], ibias = input_b[v];
    const float cden0 = cmt0 + gl0 + EPS_, cden1 = cmt1 + gl1 + EPS_;

    float v0 = 0.0f, v1 = 0.0f;
    float* myV = sV + wid * U_;
    myV[k0] = 0.0f;
    myV[k1] = 0.0f;
    __syncthreads();

    const float* xb = x + ((size_t)v * B_ + b) * T_;
    for (int t = 0; t < T_; ++t) {
        const float inp = fmaf(xb[t], iw, ibias);
        // Sensory contribution (I==1): fixed over the 6 unfolds.
        float es0 = __builtin_amdgcn_exp2f(fmaf(-ssa0, inp, ssc0));
        float es1 = __builtin_amdgcn_exp2f(fmaf(-ssa1, inp, ssc1));
        float sg0 = __builtin_amdgcn_rcpf(1.0f + es0);
        float sg1 = __builtin_amdgcn_rcpf(1.0f + es1);
        const float wns0 = swe0 * sg0, wns1 = swe1 * sg1;
        const float wds0 = swp0 * sg0, wds1 = swp1 * sg1;

        for (int u = 0; u < UNFOLDS_; ++u) {
            float num0 = 0.0f, num1 = 0.0f, den0 = 0.0f, den1 = 0.0f;
#pragma unroll 8
            for (int j = 0; j < U_; ++j) {
                const float vj = myV[j];                 // LDS broadcast
                const float4 q0 = sPK4[j * U_ + k0];
                const float4 q1 = sPK4[j * U_ + k1];
                // sigmoid(sigma*(vj-mu)) = rcp(1 + exp2(sigma*mu*l2e - sigma*l2e*vj))
                float ea = __builtin_amdgcn_exp2f(fmaf(-q0.x, vj, q0.y));
                float eb = __builtin_amdgcn_exp2f(fmaf(-q1.x, vj, q1.y));
                float sa = __builtin_amdgcn_rcpf(1.0f + ea);
                float sb = __builtin_amdgcn_rcpf(1.0f + eb);
                num0 = fmaf(q0.z, sa, num0);
                den0 = fmaf(q0.w, sa, den0);
                num1 = fmaf(q1.z, sb, num1);
                den1 = fmaf(q1.w, sb, den1);
            }
            v0 = (fmaf(cmt0, v0, glv0) + num0 + wns0) / (cden0 + den0 + wds0);
            v1 = (fmaf(cmt1, v1, glv1) + num1 + wns1) / (cden1 + den1 + wds1);
            myV[k0] = v0;   // same-wave LDS: stays in order, no barrier needed
            myV[k1] = v1;
        }
    }

    feats[(size_t)b * VU + p0] = fmaf(v0, output_w[p0], output_b[p0]);
    feats[(size_t)b * VU + p1] = fmaf(v1, output_w[p1], output_b[p1]);
}

// ---------------------------------------------------------------------------
// GEMM1: h = relu(feats[512x512] @ W1[512x256] + b1), fp32 WMMA 16x16x4.
// One wave per 16x16 output tile; 512 tiles; 128 WMMAs per wave over K=512.
// A frag (16x4): lane<16 -> K {k,k+1}, lane>=16 -> K {k+2,k+3}; M = lane%16.
// B frag (4x16): same K split; N = lane%16. C/D: VGPR r -> M = r + 8*(lane/16).
// ---------------------------------------------------------------------------
__global__ void __launch_bounds__(256, 1)
gemm1_wmma(const float* __restrict__ A,   // feats [512][512]
           const float* __restrict__ Bm,  // W1 [512][256]
           const float* __restrict__ bias,// b1 [256]
           float* __restrict__ Hout)      // [512][256]
{
    const int wid  = threadIdx.x >> 5;
    const int lane = threadIdx.x & 31;
    const int wg   = blockIdx.x * 8 + wid;     // 0..511 tiles
    const int m0   = (wg >> 4) * 16;
    const int n0   = (wg & 15) * 16;
    const int half = lane >> 4;
    const int lr   = lane & 15;

    v8f_t acc = {};
    const float* arow = A + (size_t)(m0 + lr) * (V_ * U_);
    const float* bcol = Bm + n0 + lr;
    for (int k = 0; k < V_ * U_; k += 4) {
        const int ka = k + 2 * half;
        v2f_t af, bf;
        af.x = arow[ka];
        af.y = arow[ka + 1];
        bf.x = bcol[(size_t)ka * H_];
        bf.y = bcol[(size_t)(ka + 1) * H_];
        acc = __builtin_amdgcn_wmma_f32_16x16x4_f32(
            false, af, false, bf, (short)0, acc, false, false);
    }
    const float bv = bias[n0 + lr];
#pragma unroll
    for (int r = 0; r < 8; ++r) {
        const int row = m0 + r + 8 * half;
        float val = acc[r] + bv;
        Hout[(size_t)row * H_ + (n0 + lr)] = val > 0.0f ? val : 0.0f;
    }
}

// ---------------------------------------------------------------------------
// GEMM2: out[512x10] = h @ W2 + b2. Tiny (1.3 MFLOP): plain per-thread dot.
// ---------------------------------------------------------------------------
__global__ void gemm2_small(const float* __restrict__ Hin,
                            const float* __restrict__ W2,
                            const float* __restrict__ b2,
                            float* __restrict__ out)
{
    int idx = blockIdx.x * blockDim.x + threadIdx.x;
    if (idx >= B_ * C_) return;
    int b = idx / C_, c = idx % C_;
    float acc = b2[c];
    const float* hr = Hin + (size_t)b * H_;
#pragma unroll 8
    for (int i = 0; i < H_; ++i) acc = fmaf(hr[i], W2[i * C_ + c], acc);
    out[idx] = acc;
}

// ---------------------------------------------------------------------------
extern "C" void kernel_launch(void* const* d_in, const int* in_sizes, int n_in,
                              void* d_out, int out_size, void* d_ws, size_t ws_size,
                              hipStream_t stream) {
    const float* x       = (const float*)d_in[0];
    const float* gleak   = (const float*)d_in[1];
    const float* vleak   = (const float*)d_in[2];
    const float* cm      = (const float*)d_in[3];
    const float* w       = (const float*)d_in[4];
    const float* sigma   = (const float*)d_in[5];
    const float* mu      = (const float*)d_in[6];
    const float* erev    = (const float*)d_in[7];
    const float* sw      = (const float*)d_in[8];
    const float* ssig    = (const float*)d_in[9];
    const float* smu     = (const float*)d_in[10];
    const float* serev   = (const float*)d_in[11];
    const float* input_w = (const float*)d_in[12];
    const float* input_b = (const float*)d_in[13];
    const float* out_w   = (const float*)d_in[14];
    const float* out_b   = (const float*)d_in[15];
    const float* W1      = (const float*)d_in[16];
    const float* b1      = (const float*)d_in[17];
    const float* W2      = (const float*)d_in[18];
    const float* b2      = (const float*)d_in[19];

    float* ws    = (float*)d_ws;
    float* PK    = ws;                              // V*U*U*4   = 131072
    float* vecs  = PK + V_ * U_ * U_ * 4;           // 7*V*U     = 3584
    float* feats = vecs + 7 * V_ * U_;              // B*V*U     = 262144
    float* h     = feats + (size_t)B_ * V_ * U_;    // B*H       = 131072

    ltc_prep_mat<<<(V_ * U_ * U_ + 255) / 256, 256, 0, stream>>>(w, sigma, mu, erev, PK);
    ltc_prep_vec<<<(V_ * U_ + 255) / 256, 256, 0, stream>>>(
        gleak, vleak, cm, sw, ssig, smu, serev, vecs);

    const size_t lds_bytes = (U_ * U_ * 4 + 8 * U_) * sizeof(float); // 66 KB
    ltc_scan<<<V_ * (B_ / 8), 256, lds_bytes, stream>>>(
        x, PK, vecs, input_w, input_b, out_w, out_b, feats);

    gemm1_wmma<<<64, 256, 0, stream>>>(feats, W1, b1, h);
    gemm2_small<<<(B_ * C_ + 255) / 256, 256, 0, stream>>>(h, W2, b2, (float*)d_out);
}